// GraphEncoder_67680094650552
// MI455X (gfx1250) — compile-verified
//
#include <hip/hip_runtime.h>
#include <hip/hip_bf16.h>
#include <math.h>

// ---------------- problem constants (match reference) ----------------
#define NN     50000   // nodes
#define EE     800000  // random edges (self loops appended -> EE+NN total)
#define BB     500     // graphs
#define NHEAD  4
#define NFEAT  64      // per-head features
#define HW     256     // H*F hidden width
#define GEMBW  832     // 64 + 3*256 concat width
#define SLOPE  0.2f

typedef float v2f __attribute__((ext_vector_type(2)));
typedef float v8f __attribute__((ext_vector_type(8)));

// ---------------- fp32 atomic max via integer ordering trick ----------------
__device__ __forceinline__ void atomicMaxF(float* addr, float v) {
    if (v >= 0.0f) atomicMax((int*)addr, __float_as_int(v));
    else           atomicMin((unsigned int*)addr, (unsigned int)__float_as_int(v));
}

// ---------------- generic fp32 WMMA GEMM: C = act(A[M,K] @ W[K,Nout] + bias) ----
// One wave computes a 16x32 tile (two 16x16 accumulators sharing the A fragment)
// with V_WMMA_F32_16X16X4_F32. LDA/LDW are compile-time so every load in the
// unrolled K loop is an immediate offset from two stepping base pointers.
// Requirements (all call sites satisfy them): K % 16 == 0, Nout % 32 == 0.
// M arbitrary (row-clamped loads, guarded stores). Inner loop is branch-free:
// EXEC is all-ones at every WMMA. Block = 256 threads = 8 waves.
template<int LDA, int LDW>
__global__ void wmma_gemm_f32(const float* __restrict__ A,
                              const float* __restrict__ W,
                              const float* __restrict__ bias,
                              float* __restrict__ C, int ldc,
                              int M, int K, int Nout, int do_relu)
{
    const int lane = threadIdx.x & 31;
    const int wave = threadIdx.x >> 5;
    const int row0 = blockIdx.x << 4;
    const int col0 = ((blockIdx.y << 3) + wave) << 5;    // 32 cols per wave
    if (col0 >= Nout) return;                            // wave-uniform

    const int half = lane >> 4;                          // 0: lanes 0-15, 1: 16-31
    const int l16  = lane & 15;
    int arow = row0 + l16;                               // A fragment row
    const bool row_ok = (arow < M);
    if (!row_ok) arow = M - 1;                           // clamp: loads stay legal
    const int bcol = col0 + l16;                         // B fragment col (tile 0)

    const float* ap = A + (size_t)arow * LDA + (half << 1);
    const float* bp = W + (size_t)(half << 1) * LDW + bcol;

    v8f acc0 = {}, acc1 = {};
    for (int k = 0; k < K; k += 16) {
#pragma unroll
        for (int u = 0; u < 2; ++u) {                    // two K-step-8 groups
            const v2f a0 = *(const v2f*)(ap + 8 * u);
            const v2f a1 = *(const v2f*)(ap + 8 * u + 4);
            v2f b00, b01, b10, b11;
            const float* b = bp + 8 * u * LDW;
            b00.x = b[0];             b00.y = b[LDW];
            b01.x = b[16];            b01.y = b[LDW + 16];
            b10.x = b[4 * LDW];       b10.y = b[5 * LDW];
            b11.x = b[4 * LDW + 16];  b11.y = b[5 * LDW + 16];
            acc0 = __builtin_amdgcn_wmma_f32_16x16x4_f32(false, a0, false, b00, (short)0, acc0, false, false);
            acc1 = __builtin_amdgcn_wmma_f32_16x16x4_f32(false, a0, false, b01, (short)0, acc1, false, false);
            acc0 = __builtin_amdgcn_wmma_f32_16x16x4_f32(false, a1, false, b10, (short)0, acc0, false, false);
            acc1 = __builtin_amdgcn_wmma_f32_16x16x4_f32(false, a1, false, b11, (short)0, acc1, false, false);
        }
        ap += 16;
        bp += 16 * LDW;
    }

#pragma unroll
    for (int r = 0; r < 8; ++r) {                        // D: vgpr r -> row r + 8*half
        const int rr = row0 + r + (half << 3);
        if (rr < M) {
            float v0 = acc0[r], v1 = acc1[r];
            if (bias)    { v0 += bias[bcol]; v1 += bias[bcol + 16]; }
            if (do_relu) { v0 = fmaxf(v0, 0.0f); v1 = fmaxf(v1, 0.0f); }
            C[(size_t)rr * ldc + bcol]      = v0;
            C[(size_t)rr * ldc + bcol + 16] = v1;
        }
    }
}

// ---------------- node featurization: xm = [relu(nf@bboxW+b), relu(embed[mask])] --
__global__ void featurize(const float* __restrict__ nf, const int* __restrict__ mask,
                          const float* __restrict__ bW, const float* __restrict__ bb,
                          const float* __restrict__ me, float* __restrict__ xm)
{
    int gid = blockIdx.x * blockDim.x + threadIdx.x;
    if (gid >= NN * 128) return;
    int n = gid >> 7, j = gid & 127;
    float v;
    if (j < 64) {
        float acc = bb[j];
#pragma unroll
        for (int i = 0; i < 5; ++i) acc += nf[n * 5 + i] * bW[i * 64 + j];
        v = acc;
    } else {
        v = me[mask[n] * 64 + (j - 64)];
    }
    xm[(size_t)n * 128 + j] = fmaxf(v, 0.0f);
}

// ---------------- per-(node,head) attention logits + per-layer buffer init -------
__global__ void attn_prep(const float* __restrict__ h,
                          const float* __restrict__ att_s, const float* __restrict__ att_d,
                          float* __restrict__ a_s, float* __restrict__ a_d,
                          float* __restrict__ emax, float* __restrict__ denom,
                          float* __restrict__ outb)
{
    int gid = blockIdx.x * blockDim.x + threadIdx.x;   // n*4 + head
    if (gid >= NN * NHEAD) return;
    int n = gid >> 2, hd = gid & 3;
    const float* hp = h + (size_t)n * HW + hd * NFEAT;
    const float* sp = att_s + hd * NFEAT;
    const float* dp = att_d + hd * NFEAT;
    float ss = 0.0f, dd = 0.0f;
#pragma unroll 8
    for (int f = 0; f < NFEAT; ++f) { float v = hp[f]; ss += v * sp[f]; dd += v * dp[f]; }
    a_s[gid] = ss; a_d[gid] = dd;
    emax[gid] = -INFINITY; denom[gid] = 0.0f;
    float* op = outb + (size_t)n * HW + hd * NFEAT;
#pragma unroll 8
    for (int f = 0; f < NFEAT; ++f) op[f] = 0.0f;
}

__device__ __forceinline__ void edge_ends(const int* ei, int e, int& s, int& d) {
    if (e < EE) { s = ei[e]; d = ei[EE + e]; }
    else        { s = e - EE; d = e - EE; }            // appended self loops
}

// ---------------- softmax pass 1: segment max over dst --------------------------
__global__ void edge_max(const int* __restrict__ ei,
                         const float* __restrict__ a_s, const float* __restrict__ a_d,
                         float* __restrict__ emax)
{
    int gid = blockIdx.x * blockDim.x + threadIdx.x;   // e*4 + head
    if (gid >= (EE + NN) * NHEAD) return;
    int e = gid >> 2, hd = gid & 3;
    int s, d; edge_ends(ei, e, s, d);
    float x = a_s[s * 4 + hd] + a_d[d * 4 + hd];
    x = (x > 0.0f) ? x : SLOPE * x;
    atomicMaxF(&emax[d * 4 + hd], x);
}

// ---------------- softmax pass 2: segment sum of exp ----------------------------
__global__ void edge_sum(const int* __restrict__ ei,
                         const float* __restrict__ a_s, const float* __restrict__ a_d,
                         const float* __restrict__ emax, float* __restrict__ denom)
{
    int gid = blockIdx.x * blockDim.x + threadIdx.x;
    if (gid >= (EE + NN) * NHEAD) return;
    int e = gid >> 2, hd = gid & 3;
    int s, d; edge_ends(ei, e, s, d);
    float x = a_s[s * 4 + hd] + a_d[d * 4 + hd];
    x = (x > 0.0f) ? x : SLOPE * x;
    unsafeAtomicAdd(&denom[d * 4 + hd], __expf(x - emax[d * 4 + hd]));
}

// ---------------- weighted scatter: out[dst] += alpha * h[src] ------------------
// One thread per (edge, 4-feature chunk): float4 gather + 4 native fp atomics.
__global__ void edge_agg(const int* __restrict__ ei,
                         const float* __restrict__ a_s, const float* __restrict__ a_d,
                         const float* __restrict__ emax, const float* __restrict__ denom,
                         const float* __restrict__ h, float* __restrict__ outb)
{
    long long gid = (long long)blockIdx.x * blockDim.x + threadIdx.x;
    if (gid >= (long long)(EE + NN) * 64) return;
    int e  = (int)(gid >> 6);
    int f0 = ((int)gid & 63) << 2;                     // 0..252
    int hd = f0 >> 6;
    int s, d; edge_ends(ei, e, s, d);
    float x = a_s[s * 4 + hd] + a_d[d * 4 + hd];
    x = (x > 0.0f) ? x : SLOPE * x;
    float alpha = __expf(x - emax[d * 4 + hd]) / denom[d * 4 + hd];
    const float4 hv = *(const float4*)(h + (size_t)s * HW + f0);
    float* op = outb + (size_t)d * HW + f0;
    unsafeAtomicAdd(op + 0, alpha * hv.x);
    unsafeAtomicAdd(op + 1, alpha * hv.y);
    unsafeAtomicAdd(op + 2, alpha * hv.z);
    unsafeAtomicAdd(op + 3, alpha * hv.w);
}

// ---------------- bias + relu epilogue ------------------------------------------
__global__ void bias_relu(float* __restrict__ x, const float* __restrict__ b)
{
    long long gid = (long long)blockIdx.x * blockDim.x + threadIdx.x;
    if (gid >= (long long)NN * HW) return;
    x[gid] = fmaxf(x[gid] + b[(int)gid & (HW - 1)], 0.0f);
}

// ---------------- segment max into graph embedding (values >= 0 post-relu) ------
__global__ void seg_max(const float* __restrict__ x, const int* __restrict__ batch,
                        float* __restrict__ g, int width, int goff)
{
    long long gid = (long long)blockIdx.x * blockDim.x + threadIdx.x;
    if (gid >= (long long)NN * width) return;
    int n = (int)(gid / width), c = (int)(gid % width);
    float v = x[(size_t)n * width + c];                // post-relu: v >= 0
    atomicMax((int*)&g[(size_t)batch[n] * GEMBW + goff + c], __float_as_int(v));
}

__global__ void fill_zero(float* __restrict__ p, long long n)
{
    long long gid = (long long)blockIdx.x * blockDim.x + threadIdx.x;
    if (gid < n) p[gid] = 0.0f;
}

// =================================================================================
extern "C" void kernel_launch(void* const* d_in, const int* in_sizes, int n_in,
                              void* d_out, int out_size, void* d_ws, size_t ws_size,
                              hipStream_t stream)
{
    const float* nf     = (const float*)d_in[0];
    const int*   mask   = (const int*)  d_in[1];
    const int*   batch  = (const int*)  d_in[2];
    const int*   ei     = (const int*)  d_in[3];
    const float* bboxW  = (const float*)d_in[4];
    const float* bboxb  = (const float*)d_in[5];
    const float* membed = (const float*)d_in[6];
    const float* nodeW  = (const float*)d_in[7];
    const float* nodeb  = (const float*)d_in[8];
    const float* Wl[3]  = {(const float*)d_in[9],  (const float*)d_in[13], (const float*)d_in[17]};
    const float* asl[3] = {(const float*)d_in[10], (const float*)d_in[14], (const float*)d_in[18]};
    const float* adl[3] = {(const float*)d_in[11], (const float*)d_in[15], (const float*)d_in[19]};
    const float* bl[3]  = {(const float*)d_in[12], (const float*)d_in[16], (const float*)d_in[20]};
    const float* aggW   = (const float*)d_in[21];
    const float* aggb   = (const float*)d_in[22];
    const float* muW    = (const float*)d_in[23];
    const float* mub    = (const float*)d_in[24];
    const float* varW   = (const float*)d_in[25];
    const float* varb   = (const float*)d_in[26];
    float* out = (float*)d_out;

    // ---- workspace carving (floats) ----
    float* ws = (float*)d_ws;
    size_t o = 0;
    float* xm     = ws + o; o += (size_t)NN * 128;
    float* n0     = ws + o; o += (size_t)NN * 64;
    float* hbuf   = ws + o; o += (size_t)NN * HW;
    float* ob0    = ws + o; o += (size_t)NN * HW;
    float* ob1    = ws + o; o += (size_t)NN * HW;
    float* a_s    = ws + o; o += (size_t)NN * NHEAD;
    float* a_d    = ws + o; o += (size_t)NN * NHEAD;
    float* emax   = ws + o; o += (size_t)NN * NHEAD;
    float* denom  = ws + o; o += (size_t)NN * NHEAD;
    float* gemb   = ws + o; o += (size_t)BB * GEMBW;
    float* latent = ws + o; o += (size_t)BB * HW;

    const int TPB = 256;
    auto nblk = [](long long n, int t) { return (unsigned)((n + t - 1) / t); };

    // ---- featurization: xm [N,128], then n0 = relu(xm @ nodeW + nodeb) [N,64] ----
    featurize<<<nblk((long long)NN * 128, TPB), TPB, 0, stream>>>(nf, mask, bboxW, bboxb, membed, xm);
    {
        dim3 g((NN + 15) / 16, 1);
        wmma_gemm_f32<128, 64><<<g, TPB, 0, stream>>>(xm, nodeW, nodeb, n0, 64, NN, 128, 64, 1);
    }

    // ---- graph embedding buffer + first segment-max (n0, width 64, offset 0) ----
    fill_zero<<<nblk((long long)BB * GEMBW, TPB), TPB, 0, stream>>>(gemb, (long long)BB * GEMBW);
    seg_max<<<nblk((long long)NN * 64, TPB), TPB, 0, stream>>>(n0, batch, gemb, 64, 0);

    // ---- three GAT layers ----
    const float* lin[3]  = {n0, ob0, ob1};
    float*       lout[3] = {ob0, ob1, ob0};
    const long long eh   = (long long)(EE + NN) * NHEAD;
    const long long ef   = (long long)(EE + NN) * 64;
    dim3 gN((NN + 15) / 16, 1);

    for (int l = 0; l < 3; ++l) {
        // h = in @ W   [N, Kin] x [Kin, 256]
        if (l == 0)
            wmma_gemm_f32<64, 256><<<gN, TPB, 0, stream>>>(lin[l], Wl[l], nullptr, hbuf, HW, NN, 64, HW, 0);
        else
            wmma_gemm_f32<256, 256><<<gN, TPB, 0, stream>>>(lin[l], Wl[l], nullptr, hbuf, HW, NN, 256, HW, 0);
        attn_prep<<<nblk((long long)NN * NHEAD, TPB), TPB, 0, stream>>>(
            hbuf, asl[l], adl[l], a_s, a_d, emax, denom, lout[l]);
        edge_max<<<nblk(eh, TPB), TPB, 0, stream>>>(ei, a_s, a_d, emax);
        edge_sum<<<nblk(eh, TPB), TPB, 0, stream>>>(ei, a_s, a_d, emax, denom);
        edge_agg<<<nblk(ef, TPB), TPB, 0, stream>>>(ei, a_s, a_d, emax, denom, hbuf, lout[l]);
        bias_relu<<<nblk((long long)NN * HW, TPB), TPB, 0, stream>>>(lout[l], bl[l]);
        seg_max<<<nblk((long long)NN * HW, TPB), TPB, 0, stream>>>(lout[l], batch, gemb, HW, 64 + l * HW);
    }

    // ---- latent = gemb @ aggW + aggb ; mu / log_var -> d_out ----
    {
        dim3 g((BB + 15) / 16, 1);
        wmma_gemm_f32<832, 256><<<g, TPB, 0, stream>>>(gemb, aggW, aggb, latent, HW, BB, GEMBW, HW, 0);
        wmma_gemm_f32<256, 256><<<g, TPB, 0, stream>>>(latent, muW,  mub,  out,                   HW, BB, HW, HW, 0);
        wmma_gemm_f32<256, 256><<<g, TPB, 0, stream>>>(latent, varW, varb, out + (size_t)BB * HW, HW, BB, HW, HW, 0);
    }
}